// MyGCN_67164698575202
// MI455X (gfx1250) — compile-verified
//
#include <hip/hip_runtime.h>
#include <hip/hip_bf16.h>

#define N_NODES 50000
#define N_EDGES 800000
#define D 128
#define LDA (D + 4)   // LDS pad: stride 132 dwords -> 4-bank skew per row, no conflicts

typedef float v2f __attribute__((ext_vector_type(2)));
typedef float v8f __attribute__((ext_vector_type(8)));

// ---------------------------------------------------------------- utilities
__global__ void gcn_fill_f32(float* __restrict__ p, float v, int n) {
    int i = blockIdx.x * blockDim.x + threadIdx.x;
    if (i < n) p[i] = v;
}

// deg was pre-filled with 1.0 (self-loop); add 1 per incoming edge.
__global__ void gcn_degree(const long long* __restrict__ dst, float* __restrict__ deg) {
    int i = blockIdx.x * blockDim.x + threadIdx.x;
    if (i < N_EDGES) atomicAdd(&deg[(int)dst[i]], 1.0f);
}

__global__ void gcn_rsqrt(float* __restrict__ deg, int n) {
    int i = blockIdx.x * blockDim.x + threadIdx.x;
    if (i < n) {
        float d = deg[i];
        deg[i] = (d > 0.0f) ? rsqrtf(d) : 0.0f;
    }
}

// ------------------------------------------------------- WMMA fp32 GEMM
// H[N,128] = X[N,128] @ W[128,128]
// grid.x = 3125 (16-row tiles), block = 256 threads = 8 waves,
// wave w computes the 16x16 output tile at columns [16w, 16w+16).
__global__ __launch_bounds__(256) void gcn_gemm_wmma(const float* __restrict__ X,
                                                     const float* __restrict__ W,
                                                     float* __restrict__ H) {
    __shared__ float As[16 * LDA];

    const int tid  = threadIdx.x;
    const int wave = tid >> 5;
    const int lane = tid & 31;
    const int m0   = blockIdx.x * 16;

    // Cooperative stage of the 16x128 A panel (512 float4, 2 per thread).
    {
        const float4* src = (const float4*)(X + (size_t)m0 * D);
#pragma unroll
        for (int it = 0; it < 2; ++it) {
            int idx = tid + it * 256;          // 0..511
            int r = idx >> 5;                  // row 0..15
            int c = idx & 31;                  // float4 col 0..31
            *(float4*)&As[r * LDA + c * 4] = src[r * (D / 4) + c];
        }
    }
    __syncthreads();

    const int n0    = wave * 16;
    const int rc    = lane & 15;          // A: row M;  B/C: column N
    const int khalf = (lane >> 4) << 1;   // lanes 0-15 -> K={0,1}, lanes 16-31 -> K={2,3}

    v8f acc = {};
#pragma unroll
    for (int k = 0; k < D; k += 4) {
        v2f a, b;
        // 32-bit A 16x4 fragment (ISA 7.12.2): VGPR0=K=khalf, VGPR1=K=khalf+1
        a.x = As[rc * LDA + k + khalf];
        a.y = As[rc * LDA + k + khalf + 1];
        // 32-bit B 4x16 fragment: row K striped across lanes (coalesced per half-wave)
        b.x = W[(size_t)(k + khalf) * D + n0 + rc];
        b.y = W[(size_t)(k + khalf + 1) * D + n0 + rc];
        acc = __builtin_amdgcn_wmma_f32_16x16x4_f32(false, a, false, b,
                                                    (short)0, acc, false, false);
    }

    // C/D layout: VGPR v -> M = (lane>=16 ? 8 : 0) + v, N = lane&15
    const int mb = (lane >> 4) * 8;
#pragma unroll
    for (int v = 0; v < 8; ++v)
        H[(size_t)(m0 + mb + v) * D + n0 + rc] = acc[v];
}

// ------------------------------------------------- edge gather + scatter-add
// 32 lanes per edge, float4 per lane: agg[dst] += h[src] * dis[src]*dis[dst]
__global__ __launch_bounds__(256) void gcn_aggregate(const float* __restrict__ H,
                                                     const long long* __restrict__ src,
                                                     const long long* __restrict__ dst,
                                                     const float* __restrict__ dis,
                                                     float* __restrict__ agg) {
    long long g = (long long)blockIdx.x * blockDim.x + threadIdx.x;
    int e = (int)(g >> 5);
    if (e >= N_EDGES) return;
    int c = ((int)g & 31) * 4;

    int s = (int)src[e];
    int d = (int)dst[e];
    float norm = dis[s] * dis[d];

    float4 hv = *(const float4*)(H + (size_t)s * D + c);
    float* out = agg + (size_t)d * D + c;
    atomicAdd(out + 0, hv.x * norm);
    atomicAdd(out + 1, hv.y * norm);
    atomicAdd(out + 2, hv.z * norm);
    atomicAdd(out + 3, hv.w * norm);
}

// ------------------------- epilogue: self-loop term + bias + ReLU (in place ok)
__global__ void gcn_finalize(const float* __restrict__ agg, const float* __restrict__ H,
                             const float* __restrict__ dis, const float* __restrict__ b,
                             float* __restrict__ out) {
    int i = blockIdx.x * blockDim.x + threadIdx.x;
    if (i >= N_NODES * D) return;
    int row = i >> 7;
    int col = i & (D - 1);
    float di = dis[row];
    float v = agg[i] + H[i] * di * di + b[col];
    out[i] = v > 0.0f ? v : 0.0f;
}

// ---------------------------------------------------------------- launcher
extern "C" void kernel_launch(void* const* d_in, const int* in_sizes, int n_in,
                              void* d_out, int out_size, void* d_ws, size_t ws_size,
                              hipStream_t stream) {
    const float* x            = (const float*)d_in[0];
    const long long* edge     = (const long long*)d_in[1];  // int64 per reference
    const float* W1           = (const float*)d_in[2];
    const float* b1           = (const float*)d_in[3];
    const float* W2           = (const float*)d_in[4];
    const float* b2           = (const float*)d_in[5];
    float* out                = (float*)d_out;

    const long long* esrc = edge;            // edge_index[0]
    const long long* edst = edge + N_EDGES;  // edge_index[1]

    // Workspace layout (floats): dis[50176] | h[N*D] | agg[N*D]
    float* dis = (float*)d_ws;
    float* h   = dis + 50176;                         // 256-aligned
    float* agg = h + (size_t)N_NODES * D;

    const int B = 256;
    const int nBlkN   = (N_NODES + B - 1) / B;
    const int nBlkND  = (N_NODES * D) / B;            // 25000 exactly
    const int nBlkE   = (N_EDGES + B - 1) / B;
    const int nBlkAgg = (N_EDGES * 32) / B;           // 100000 exactly
    const int nBlkGemm = N_NODES / 16;                // 3125 exactly

    // Degrees (self-loop contributes the initial 1.0) -> dis = rsqrt(deg)
    gcn_fill_f32<<<nBlkN, B, 0, stream>>>(dis, 1.0f, N_NODES);
    gcn_degree<<<nBlkE, B, 0, stream>>>(edst, dis);
    gcn_rsqrt<<<nBlkN, B, 0, stream>>>(dis, N_NODES);

    // ---- layer 1
    gcn_gemm_wmma<<<nBlkGemm, B, 0, stream>>>(x, W1, h);
    gcn_fill_f32<<<nBlkND, B, 0, stream>>>(agg, 0.0f, N_NODES * D);
    gcn_aggregate<<<nBlkAgg, B, 0, stream>>>(h, esrc, edst, dis, agg);
    gcn_finalize<<<nBlkND, B, 0, stream>>>(agg, h, dis, b1, agg);   // agg <- relu(layer1)

    // ---- layer 2
    gcn_gemm_wmma<<<nBlkGemm, B, 0, stream>>>(agg, W2, h);
    gcn_fill_f32<<<nBlkND, B, 0, stream>>>(out, 0.0f, N_NODES * D);
    gcn_aggregate<<<nBlkAgg, B, 0, stream>>>(h, esrc, edst, dis, out);
    gcn_finalize<<<nBlkND, B, 0, stream>>>(out, h, dis, b2, out);
}